// PRMPConv_1099511628124
// MI455X (gfx1250) — compile-verified
//
#include <hip/hip_runtime.h>
#include <hip/hip_bf16.h>

#define NN     50000
#define EE     800000
#define D      128
#define LDSS   132      // padded row stride (floats): 132 % 64 = 4 -> conflict-free column reads
#define MT     5        // 16-row WMMA tiles per block (50000 = 625 * 80, exact)
#define ROWS   (MT*16)  // 80 rows per block
#define LN_EPS 1e-5f

typedef __attribute__((ext_vector_type(2))) float v2f;
typedef __attribute__((ext_vector_type(8))) float v8f;

// CDNA5 fp32 matrix core: D(16x16,f32) = A(16x4,f32) x B(4x16,f32) + C
__device__ __forceinline__ v8f wmma4(v2f a, v2f b, v8f c) {
  return __builtin_amdgcn_wmma_f32_16x16x4_f32(false, a, false, b, (short)0, c,
                                               false, false);
}

// ---------------------------------------------------------------------------
// Zero the normed-residual accumulators (d_out, reused as agg) and counts.
// ---------------------------------------------------------------------------
__global__ void zero_kernel(float* __restrict__ a, int na,
                            float* __restrict__ b, int nb) {
  int i = blockIdx.x * blockDim.x + threadIdx.x;
  int total = na + nb;
  int stride = gridDim.x * blockDim.x;
  for (; i < total; i += stride) {
    if (i < na) a[i] = 0.0f;
    else        b[i - na] = 0.0f;
  }
}

// ---------------------------------------------------------------------------
// Kernel A: per-node predictive MLP  pred = relu(x_dst@W1+b1)@W2+b2
// 80 nodes/block, 8 waves; each wave owns one 16-col tile across 5 M-tiles,
// so every B-fragment load feeds 5 v_wmma ops (5x weight-traffic reuse).
// ---------------------------------------------------------------------------
__global__ __launch_bounds__(256) void node_mlp_kernel(
    const float* __restrict__ x_dst,
    const float* __restrict__ W1, const float* __restrict__ b1,
    const float* __restrict__ W2, const float* __restrict__ b2,
    float* __restrict__ pred) {
  __shared__ float xs[ROWS * LDSS];
  __shared__ float hs[ROWS * LDSS];

  const int t     = threadIdx.x;
  const int node0 = blockIdx.x * ROWS;
  const int c16   = t & 15;

  // Stage the 80x128 input tile (each thread: 5 rows x 8 contiguous floats).
#pragma unroll
  for (int j = 0; j < MT; ++j) {
    int row = (t >> 4) + 16 * j;
    const float* src = x_dst + (size_t)(node0 + row) * D + c16 * 8;
    float4 v0 = *(const float4*)(src);
    float4 v1 = *(const float4*)(src + 4);
    *(float4*)&xs[row * LDSS + c16 * 8]     = v0;
    *(float4*)&xs[row * LDSS + c16 * 8 + 4] = v1;
  }
  __syncthreads();

  const int lane = t & 31;
  const int wv   = t >> 5;       // 0..7 -> output column tile
  const int half = lane >> 4;    // K sub-select per ISA A/B layout
  const int l16  = lane & 15;
  const int n0   = wv * 16;

  v8f acc[MT];

  // ---- GEMM 1: h = relu(x @ W1 + b1), bias folded into C fragments ----
  {
    float bias = b1[n0 + l16];
#pragma unroll
    for (int m = 0; m < MT; ++m)
#pragma unroll
      for (int i = 0; i < 8; ++i) acc[m][i] = bias;
  }
#pragma unroll 4
  for (int k0 = 0; k0 < D; k0 += 4) {
    v2f b;
    b.x = W1[(k0 + 2 * half) * D + n0 + l16];
    b.y = W1[(k0 + 2 * half + 1) * D + n0 + l16];
#pragma unroll
    for (int m = 0; m < MT; ++m) {
      v2f a = *(const v2f*)&xs[(m * 16 + l16) * LDSS + k0 + 2 * half];
      acc[m] = wmma4(a, b, acc[m]);
    }
  }
#pragma unroll
  for (int m = 0; m < MT; ++m)
#pragma unroll
    for (int i = 0; i < 8; ++i) {
      float v = acc[m][i] > 0.0f ? acc[m][i] : 0.0f;
      hs[(m * 16 + i + 8 * half) * LDSS + n0 + l16] = v;
    }
  __syncthreads();

  // ---- GEMM 2: pred = h @ W2 + b2 ----
  {
    float bias = b2[n0 + l16];
#pragma unroll
    for (int m = 0; m < MT; ++m)
#pragma unroll
      for (int i = 0; i < 8; ++i) acc[m][i] = bias;
  }
#pragma unroll 4
  for (int k0 = 0; k0 < D; k0 += 4) {
    v2f b;
    b.x = W2[(k0 + 2 * half) * D + n0 + l16];
    b.y = W2[(k0 + 2 * half + 1) * D + n0 + l16];
#pragma unroll
    for (int m = 0; m < MT; ++m) {
      v2f a = *(const v2f*)&hs[(m * 16 + l16) * LDSS + k0 + 2 * half];
      acc[m] = wmma4(a, b, acc[m]);
    }
  }
#pragma unroll
  for (int m = 0; m < MT; ++m)
#pragma unroll
    for (int i = 0; i < 8; ++i)
      pred[(size_t)(node0 + m * 16 + i + 8 * half) * D + n0 + l16] = acc[m][i];
}

// ---------------------------------------------------------------------------
// Kernel B: per-edge  gather -> residual -> LayerNorm -> atomic scatter of the
// NORMED vector (the @Wl GEMM is hoisted to node level by linearity).
// Pure L2-gather/atomic kernel: both tables (25.6MB) are L2-resident.
// ---------------------------------------------------------------------------
__global__ __launch_bounds__(256) void edge_scatter_kernel(
    const float* __restrict__ x_src,
    const float* __restrict__ pred,
    const long long* __restrict__ src_idx,
    const long long* __restrict__ dst_idx,
    const float* __restrict__ gamma, const float* __restrict__ beta,
    float* __restrict__ agg, float* __restrict__ cnt) {
  __shared__ float psum[16 * 16];
  __shared__ float psq[16 * 16];
  __shared__ int   sid[16];
  __shared__ int   did[16];

  const int t  = threadIdx.x;
  const long long e0 = (long long)blockIdx.x * 16;   // E = 50000 * 16, exact

  if (t < 16) {
    int s = (int)src_idx[e0 + t];
    int d = (int)dst_idx[e0 + t];
    sid[t] = s;
    did[t] = d;
    atomicAdd(&cnt[d], 1.0f);                        // scatter_mean count
    __builtin_prefetch(x_src + (size_t)s * D, 0, 3); // warm the gather rows
    __builtin_prefetch(pred + (size_t)d * D, 0, 3);
  }
  __syncthreads();

  const int row = t >> 4;
  const int c16 = t & 15;

  // residual = x_src[src] - pred[dst], registers + partial LN stats
  const float* xr = x_src + (size_t)sid[row] * D + c16 * 8;
  const float* pr = pred + (size_t)did[row] * D + c16 * 8;
  float r[8];
  float s = 0.0f, sq = 0.0f;
#pragma unroll
  for (int i = 0; i < 8; ++i) {
    float v = xr[i] - pr[i];
    r[i] = v;
    s += v;
    sq += v * v;
  }
  psum[t] = s;
  psq[t]  = sq;
  __syncthreads();

  float tot = 0.0f, totq = 0.0f;
#pragma unroll
  for (int i = 0; i < 16; ++i) {
    tot  += psum[row * 16 + i];
    totq += psq[row * 16 + i];
  }
  const float mu   = tot * (1.0f / 128.0f);
  const float var  = totq * (1.0f / 128.0f) - mu * mu;
  const float rstd = __frsqrt_rn(var + LN_EPS);

  float* dstp = agg + (size_t)did[row] * D + c16 * 8;
#pragma unroll
  for (int i = 0; i < 8; ++i) {
    int col = c16 * 8 + i;
    atomicAdd(&dstp[i], (r[i] - mu) * rstd * gamma[col] + beta[col]);
  }
}

// ---------------------------------------------------------------------------
// Kernel C: out = (agg @ Wl + cnt*bl) / max(cnt, 1)   (in place on d_out;
// the agg tile is staged to LDS before being overwritten).
// ---------------------------------------------------------------------------
__global__ __launch_bounds__(256) void node_out_kernel(
    const float* __restrict__ Wl, const float* __restrict__ bl,
    const float* __restrict__ cnt,
    float* __restrict__ out) {                        // in: agg, out: result
  __shared__ float as[ROWS * LDSS];
  __shared__ float ct[ROWS];

  const int t     = threadIdx.x;
  const int node0 = blockIdx.x * ROWS;
  const int c16   = t & 15;

  if (t < ROWS) ct[t] = cnt[node0 + t];
#pragma unroll
  for (int j = 0; j < MT; ++j) {
    int row = (t >> 4) + 16 * j;
    const float* src = out + (size_t)(node0 + row) * D + c16 * 8;
    float4 v0 = *(const float4*)(src);
    float4 v1 = *(const float4*)(src + 4);
    *(float4*)&as[row * LDSS + c16 * 8]     = v0;
    *(float4*)&as[row * LDSS + c16 * 8 + 4] = v1;
  }
  __syncthreads();

  const int lane = t & 31;
  const int wv   = t >> 5;
  const int half = lane >> 4;
  const int l16  = lane & 15;
  const int n0   = wv * 16;

  v8f acc[MT];
#pragma unroll
  for (int m = 0; m < MT; ++m)
#pragma unroll
    for (int i = 0; i < 8; ++i) acc[m][i] = 0.0f;

#pragma unroll 4
  for (int k0 = 0; k0 < D; k0 += 4) {
    v2f b;
    b.x = Wl[(k0 + 2 * half) * D + n0 + l16];
    b.y = Wl[(k0 + 2 * half + 1) * D + n0 + l16];
#pragma unroll
    for (int m = 0; m < MT; ++m) {
      v2f a = *(const v2f*)&as[(m * 16 + l16) * LDSS + k0 + 2 * half];
      acc[m] = wmma4(a, b, acc[m]);
    }
  }

  const float blv = bl[n0 + l16];
#pragma unroll
  for (int m = 0; m < MT; ++m)
#pragma unroll
    for (int i = 0; i < 8; ++i) {
      int row = m * 16 + i + 8 * half;
      float c = ct[row];
      out[(size_t)(node0 + row) * D + n0 + l16] =
          (acc[m][i] + c * blv) / fmaxf(c, 1.0f);
    }
}

extern "C" void kernel_launch(void* const* d_in, const int* in_sizes, int n_in,
                              void* d_out, int out_size, void* d_ws,
                              size_t ws_size, hipStream_t stream) {
  (void)in_sizes; (void)n_in; (void)out_size; (void)ws_size;

  const float*     x_src = (const float*)d_in[0];
  const float*     x_dst = (const float*)d_in[1];
  const long long* eidx  = (const long long*)d_in[2];  // int64 [2, E]
  const float*     W1    = (const float*)d_in[3];
  const float*     b1    = (const float*)d_in[4];
  const float*     W2    = (const float*)d_in[5];
  const float*     b2    = (const float*)d_in[6];
  const float*     gamma = (const float*)d_in[7];
  const float*     beta  = (const float*)d_in[8];
  const float*     Wl    = (const float*)d_in[9];
  const float*     bl    = (const float*)d_in[10];

  float* out  = (float*)d_out;                 // doubles as normed-sum agg
  float* pred = (float*)d_ws;                  // [N, 128] node MLP output
  float* cnt  = pred + (size_t)NN * D;         // [N] counts

  const long long* src_idx = eidx;
  const long long* dst_idx = eidx + EE;

  // 1) zero agg (d_out) + counts
  {
    int total = NN * D + NN;
    zero_kernel<<<(total + 255) / 256, 256, 0, stream>>>(out, NN * D, cnt, NN);
  }
  // 2) per-node MLP (16x fewer GEMM FLOPs than per-edge)
  node_mlp_kernel<<<NN / ROWS, 256, 0, stream>>>(x_dst, W1, b1, W2, b2, pred);
  // 3) per-edge gather + LN + atomic scatter of normed residuals
  edge_scatter_kernel<<<EE / 16, 256, 0, stream>>>(x_src, pred, src_idx,
                                                   dst_idx, gamma, beta, out,
                                                   cnt);
  // 4) node-level output GEMM + mean + hoisted bias (in place)
  node_out_kernel<<<NN / ROWS, 256, 0, stream>>>(Wl, bl, cnt, out);
}